// DeformableModule_6468220748305
// MI455X (gfx1250) — compile-verified
//
#include <hip/hip_runtime.h>

// Problem dimensions (fixed by the reference):
//   images : (N=4, O=8, C=32, H=128, W=128) fp32
//   offsets: (N=4, 2*O=16, H=128, W=128)    fp32
//   out    : (N=4, O=8, C=32, H=128, W=128) fp32
#define NB   4
#define NO   8
#define NC   32
#define HH   128
#define WW   128
#define HW   (HH * WW)

__global__ __launch_bounds__(WW) void deformable_sample_kernel(
    const float* __restrict__ images,
    const float* __restrict__ offsets,
    float* __restrict__ out)
{
    const int w   = threadIdx.x;        // 0..127  (lane-contiguous in W)
    const int row = blockIdx.x;         // n*NO*HH + o*HH + h
    const int h   = row % HH;
    const int no  = row / HH;           // n*NO + o
    const int o   = no % NO;
    const int n   = no / NO;

    // ---- per-(n,o,h,w) offset fetch (coalesced) ----
    const int offBase = ((n * (2 * NO) + 2 * o) * HH + h) * WW + w;
    const float oh = offsets[offBase];       // off_h = offsets[n, 2o,   h, w]
    const float ow = offsets[offBase + HW];  // off_w = offsets[n, 2o+1, h, w]

    // ---- bilinear taps: computed ONCE, reused across all 32 channels ----
    const float fh  = floorf(oh);
    const float fw  = floorf(ow);
    const float frh = oh - fh;          // weight for dh=1 is frh, dh=0 is 1-frh
    const float frw = ow - fw;

    const int ih0 = (int)fh + h;
    const int iw0 = (int)fw + w;
    const int ih1 = ih0 + 1;
    const int iw1 = iw0 + 1;

    const bool vh0 = (ih0 >= 0) && (ih0 < HH);
    const bool vh1 = (ih1 >= 0) && (ih1 < HH);
    const bool vw0 = (iw0 >= 0) && (iw0 < WW);
    const bool vw1 = (iw1 >= 0) && (iw1 < WW);

    const float wh0 = 1.0f - frh, wh1 = frh;
    const float ww0 = 1.0f - frw, ww1 = frw;

    // invalid taps get weight 0 and a clamped (in-bounds) address -> no branches
    const float w00 = (vh0 && vw0) ? wh0 * ww0 : 0.0f;
    const float w01 = (vh0 && vw1) ? wh0 * ww1 : 0.0f;
    const float w10 = (vh1 && vw0) ? wh1 * ww0 : 0.0f;
    const float w11 = (vh1 && vw1) ? wh1 * ww1 : 0.0f;

    const int ihc0 = vh0 ? ih0 : 0;
    const int ihc1 = vh1 ? ih1 : 0;
    const int iwc0 = vw0 ? iw0 : 0;
    const int iwc1 = vw1 ? iw1 : 0;

    const int t00 = ihc0 * WW + iwc0;
    const int t01 = ihc0 * WW + iwc1;
    const int t10 = ihc1 * WW + iwc0;
    const int t11 = ihc1 * WW + iwc1;

    const long base = (long)no * NC * HW;   // start of images[n, o] / out[n, o]
    const float* __restrict__ img = images + base;
    float* __restrict__ op        = out + base + h * WW + w;

    // ---- channel loop: 4 loads + 4 FMA + 1 NT store per channel ----
    // unroll 8 => ~32 outstanding loads/lane (LOADcnt max 63), hides gather latency
    #pragma unroll 8
    for (int c = 0; c < NC; ++c) {
        const float* __restrict__ p = img + c * HW;
        float v = p[t00] * w00;
        v = fmaf(p[t01], w01, v);
        v = fmaf(p[t10], w10, v);
        v = fmaf(p[t11], w11, v);
        // output is write-once, never re-read by this kernel -> NT store
        // (keeps the gather-hot image lines resident in WGP$/L2)
        __builtin_nontemporal_store(v, op + c * HW);
    }
}

extern "C" void kernel_launch(void* const* d_in, const int* in_sizes, int n_in,
                              void* d_out, int out_size, void* d_ws, size_t ws_size,
                              hipStream_t stream)
{
    (void)in_sizes; (void)n_in; (void)out_size; (void)d_ws; (void)ws_size;

    const float* images  = (const float*)d_in[0];   // (4, 8, 32, 128, 128) fp32
    const float* offsets = (const float*)d_in[1];   // (4, 16, 128, 128)    fp32
    float* out           = (float*)d_out;           // (4, 8, 32, 128, 128) fp32

    const int grid = NB * NO * HH;                  // 4096 blocks, one (n,o,h) row each
    deformable_sample_kernel<<<grid, WW, 0, stream>>>(images, offsets, out);
}